// SlaterDetSampler_ordered_20366734917916
// MI455X (gfx1250) — compile-verified
//
#include <hip/hip_runtime.h>
#include <hip/hip_bf16.h>
#include <math.h>

#define DD 512
#define NN 128
#define NB 16
#define LSTR 17          // LDS row stride (coprime with 64 banks -> conflict-free)
#define TINYF 1e-30f

typedef float v2f __attribute__((ext_vector_type(2)));
typedef float v8f __attribute__((ext_vector_type(8)));

// -----------------------------------------------------------------------------
// Kernel 1: A = G = I - P P^T   (fp32 WMMA, one 16x16 tile per wave)
// Also zero-fills cond_probs and sets positions to -1.
// Grid: 128 blocks x 256 threads = 1024 waves = 1024 tiles (32x32 tile grid).
// -----------------------------------------------------------------------------
__global__ __launch_bounds__(256) void build_G_kernel(const float* __restrict__ P,
                                                      float* __restrict__ A,
                                                      float* __restrict__ out) {
  const int tid  = threadIdx.x;
  const int lane = tid & 31;
  const int gwave = blockIdx.x * 8 + (tid >> 5);
  const int m    = lane & 15;            // row (A-op) / col (B-op) within tile
  const int koff = (lane >> 4) << 1;     // K sub-offset per lane half
  const int rsel = (lane >> 4) << 3;     // C/D layout: lanes 16-31 hold M+8

  // ---- init outputs: cond_probs = 0, positions = -1 ----
  {
    const long long total = (long long)NN * DD;
    for (long long idx = (long long)blockIdx.x * 256 + tid; idx < total;
         idx += 128LL * 256)
      out[idx] = 0.0f;
    if (blockIdx.x == 0 && tid < NN) {
      int* pos = (int*)(out + (size_t)NN * DD);
      pos[tid] = -1;
    }
  }

  // ---- one 16x16 output tile per wave ----
  if (gwave < 32 * 32) {
    const int tr = (gwave >> 5) << 4;
    const int tc = (gwave & 31) << 4;
    v8f acc;
#pragma unroll
    for (int v = 0; v < 8; ++v) acc[v] = 0.0f;

    for (int kk = 0; kk < NN; kk += 4) {
      v2f a, b;
      // A-operand: 16x4 slice of P rows [tr..tr+15]
      a.x = P[(tr + m) * NN + kk + koff];
      a.y = P[(tr + m) * NN + kk + koff + 1];
      // B-operand: 4x16 slice of P^T = P rows [tc..tc+15]
      b.x = P[(tc + m) * NN + kk + koff];
      b.y = P[(tc + m) * NN + kk + koff + 1];
      acc = __builtin_amdgcn_wmma_f32_16x16x4_f32(false, a, false, b, (short)0,
                                                  acc, false, false);
    }
    const int col = tc + m;
#pragma unroll
    for (int v = 0; v < 8; ++v) {
      const int r = tr + rsel + v;
      A[r * DD + col] = ((r == col) ? 1.0f : 0.0f) - acc[v];
    }
  }
}

// -----------------------------------------------------------------------------
// Kernel 2: blocked right-looking LDL^T of (G - diag(occ)) with on-the-fly
// pivot decisions == the reference's sequential sampling scan.
// Panel width 16 (serial scalar chain); per panel both L (raw columns) and
// W = -L/pivot are staged in LDS, so the trailing WMMA update
//   A[t0.., t0..] += L * W^T
// reads BOTH operands from LDS (only the C tile streams through L2).
// One workgroup of 256 threads (8 waves).
// -----------------------------------------------------------------------------
__global__ __launch_bounds__(256) void ldl_scan_kernel(float* __restrict__ A,
                                                       const float* __restrict__ u,
                                                       float* __restrict__ out) {
  __shared__ float Lp[DD][LSTR];      // ~34 KB: raw panel columns L
  __shared__ float Wp[DD][LSTR];      // ~34 KB: W = -L / pivot
  __shared__ float s_ratio, s_cumul, s_invp;
  __shared__ int   s_k;

  const int tid  = threadIdx.x;
  const int lane = tid & 31;
  const int wave = tid >> 5;
  const int m    = lane & 15;
  const int koff = (lane >> 4) << 1;
  const int rsel = (lane >> 4) << 3;
  int* pos = (int*)(out + (size_t)NN * DD);

  if (tid == 0) { s_ratio = 1.0f; s_cumul = 0.0f; s_k = 0; s_invp = 0.0f; }
  __syncthreads();

  for (int p = 0; p < DD / NB; ++p) {
    const int j0 = p * NB;

    // ---------------- panel factorization (sequential 16 columns) -----------
    for (int ii = 0; ii < NB; ++ii) {
      const int gi = j0 + ii;

      if (tid == 0) {
        // s_i = trailing diagonal = Schur pivot of leading block
        const float s = A[gi * DD + gi];
        int   k     = s_k;
        const float ratio = s_ratio, cumul = s_cumul;
        const int   last_allowed = DD - NN + k;
        const bool  active = (k < NN) && (gi <= last_allowed);
        const float pv = active ? (-(s - 1.0f) * ratio) : 0.0f;
        int kc = k; if (kc > NN - 1) kc = NN - 1; if (kc < 0) kc = 0;
        const float uk = u[kc];
        const bool occupy =
            active && (((cumul + pv) >= uk) || (gi == last_allowed));
        float pivot = occupy ? (s - 1.0f) : s;
        if (fabsf(pivot) < TINYF) pivot = TINYF;

        if (k < NN) out[k * DD + gi] = pv;     // cond_probs[k, gi]
        if (occupy) pos[k] = gi;               // positions[k]

        s_ratio = occupy ? 1.0f : (active ? (ratio * s) : ratio);
        s_cumul = occupy ? 0.0f : (cumul + pv);
        s_k     = k + (occupy ? 1 : 0);
        s_invp  = 1.0f / pivot;
      }
      __syncthreads();

      // Stage L[:,ii] and W[:,ii] = -L[:,ii]/pivot into LDS (single L2 read)
      const float invp = s_invp;
      for (int j = j0 + tid; j < DD; j += 256) {
        const float lij = A[j * DD + gi];
        Lp[j][ii] = lij;
        Wp[j][ii] = -lij * invp;
      }
      __syncthreads();

      // rank-1 update of the remaining panel columns (L from LDS)
      for (int j = j0 + tid; j < DD; j += 256) {
        const float lij = Lp[j][ii];
        for (int c2 = ii + 1; c2 < NB; ++c2)
          A[j * DD + j0 + c2] += lij * Wp[j0 + c2][ii];
      }
      __threadfence_block();
      __syncthreads();
    }

    // ---------------- trailing update via WMMA: A += L * W^T ----------------
    const int t0 = j0 + NB;
    const int nt = (DD - t0) >> 4;         // trailing tile grid is nt x nt
    for (int t = wave; t < nt * nt; t += 8) {
      const int tr = t0 + (t / nt) * 16;
      const int tc = t0 + (t % nt) * 16;

      v8f acc;
#pragma unroll
      for (int v = 0; v < 8; ++v)          // preload C with the A tile (L2)
        acc[v] = A[(tr + rsel + v) * DD + tc + m];

#pragma unroll
      for (int kk = 0; kk < NB; kk += 4) {
        v2f a, b;
        // A-operand: L panel from LDS
        a.x = Lp[tr + m][kk + koff];
        a.y = Lp[tr + m][kk + koff + 1];
        // B-operand: W^T from LDS
        b.x = Wp[tc + m][kk + koff];
        b.y = Wp[tc + m][kk + koff + 1];
        acc = __builtin_amdgcn_wmma_f32_16x16x4_f32(false, a, false, b, (short)0,
                                                    acc, false, false);
      }
#pragma unroll
      for (int v = 0; v < 8; ++v)
        A[(tr + rsel + v) * DD + tc + m] = acc[v];
    }
    __threadfence_block();
    __syncthreads();
  }
}

// -----------------------------------------------------------------------------
extern "C" void kernel_launch(void* const* d_in, const int* in_sizes, int n_in,
                              void* d_out, int out_size, void* d_ws, size_t ws_size,
                              hipStream_t stream) {
  (void)in_sizes; (void)n_in; (void)out_size; (void)ws_size;
  const float* P = (const float*)d_in[0];   // (D, N) row-major
  const float* u = (const float*)d_in[1];   // (N,)
  float* out = (float*)d_out;               // cond_probs (N*D) then positions (N)
  float* A   = (float*)d_ws;                // D*D f32 working matrix (1 MB, L2-resident)

  build_G_kernel<<<128, 256, 0, stream>>>(P, A, out);
  ldl_scan_kernel<<<1, 256, 0, stream>>>(A, u, out);
}